// Logistic_62139586839223
// MI455X (gfx1250) — compile-verified
//
#include <hip/hip_runtime.h>
#include <hip/hip_bf16.h>

typedef __attribute__((ext_vector_type(2))) float v2f;
typedef __attribute__((ext_vector_type(8))) float v8f;

#define D784   784
#define NCOLS  7840          // 784 j-values * 10 outputs
#define NB_N   49            // 7840 / 160
#define NB_M   8             // 128 / 16
#define BLK_N  160           // 16 j * 10 o per block
#define OUT_N  1280          // 128 * 10

// y = X[128x784] @ W2mat[784x7840], fused with the j-contraction against x1.
// Each wave computes a 16(batch) x 16(col) tile with V_WMMA_F32_16X16X4_F32.
__global__ __launch_bounds__(320) void poly_gemm_partial(
    const float* __restrict__ X,      // [128, 784]
    const float* __restrict__ W,      // [615440, 10]; W2mat = W + 7840, row stride 7840
    float* __restrict__ partial)      // [49, 128, 10]
{
    __shared__ float lds[16][BLK_N];

    const int lane = threadIdx.x & 31;
    const int wave = threadIdx.x >> 5;        // 0..9
    const int half = lane >> 4;               // 0 or 1 (selects K pair)
    const int mn   = lane & 15;               // M for A-loads, N for B-loads
    const int nblk = blockIdx.x;              // 0..48
    const int m0   = blockIdx.y << 4;         // batch tile base
    const int n0   = nblk * BLK_N + wave * 16;

    const float* __restrict__ Xrow = X + (m0 + mn) * D784;       // A: row m
    const float* __restrict__ Bp   = W + NCOLS + n0 + mn;        // B: col n, row stride 7840

    v8f c = {0.f, 0.f, 0.f, 0.f, 0.f, 0.f, 0.f, 0.f};

    #pragma unroll 4
    for (int k = 0; k < D784; k += 4) {
        const int kk = k + (half << 1);
        // A 16x4 f32: lanes 0-15 hold K={k,k+1}, lanes 16-31 hold K={k+2,k+3}
        v2f a;
        a.x = Xrow[kk];
        a.y = Xrow[kk + 1];
        // B 4x16 f32: mirrored layout, N striped across lanes
        v2f b;
        b.x = Bp[kk * NCOLS];
        b.y = Bp[(kk + 1) * NCOLS];
        c = __builtin_amdgcn_wmma_f32_16x16x4_f32(
                /*neg_a=*/false, a, /*neg_b=*/false, b,
                /*c_mod=*/(short)0, c, /*reuse_a=*/false, /*reuse_b=*/false);
    }

    // Epilogue: y[b, n] * x1[b, j(n)] staged to LDS, then reduce over the 16 j's.
    const int n_local  = wave * 16 + mn;       // 0..159
    const int j_local  = n_local / 10;         // 0..15
    const int j_global = nblk * 16 + j_local;  // 0..783
    #pragma unroll
    for (int r = 0; r < 8; ++r) {
        const int m = r + (half << 3);         // D layout: M = vgpr + 8*(lane/16)
        const float xv = X[(m0 + m) * D784 + j_global];
        lds[m][n_local] = c[r] * xv;
    }
    __syncthreads();

    if (threadIdx.x < BLK_N) {                 // 16 batches x 10 outputs
        const int m = threadIdx.x / 10;
        const int o = threadIdx.x % 10;
        float s = 0.f;
        #pragma unroll
        for (int t = 0; t < 16; ++t) s += lds[m][t * 10 + o];
        partial[nblk * OUT_N + (m0 + m) * 10 + o] = s;
    }
}

// out[b,o] = bias[o] + sum_i x1[b,i]*W1[i,o] + sum_blk partial[blk,b,o]
__global__ void poly_finalize(
    const float* __restrict__ X,
    const float* __restrict__ W,
    const float* __restrict__ bias,
    const float* __restrict__ partial,
    float* __restrict__ out)
{
    const int idx = blockIdx.x * blockDim.x + threadIdx.x;
    if (idx >= OUT_N) return;
    const int b = idx / 10;
    const int o = idx % 10;

    float s = bias[o];
    const float* __restrict__ xr = X + b * D784;
    #pragma unroll 8
    for (int i = 0; i < D784; ++i)
        s += xr[i] * W[i * 10 + o];            // W1 term

    #pragma unroll
    for (int blk = 0; blk < NB_N; ++blk)
        s += partial[blk * OUT_N + idx];

    out[idx] = s;
}

extern "C" void kernel_launch(void* const* d_in, const int* in_sizes, int n_in,
                              void* d_out, int out_size, void* d_ws, size_t ws_size,
                              hipStream_t stream) {
    const float* X    = (const float*)d_in[0];   // [128, 28, 28] contiguous == [128, 784]
    const float* W    = (const float*)d_in[1];   // [615440, 10]
    const float* bias = (const float*)d_in[2];   // [10]
    float* out        = (float*)d_out;           // [128, 10]
    float* partial    = (float*)d_ws;            // needs 49*1280*4 = 250,880 B

    dim3 grid(NB_N, NB_M);                       // 49 x 8 blocks, 320 threads (10 waves)
    poly_gemm_partial<<<grid, 320, 0, stream>>>(X, W, partial);
    poly_finalize<<<(OUT_N + 255) / 256, 256, 0, stream>>>(X, W, bias, partial, out);
}